// OracleModel_85109071938307
// MI455X (gfx1250) — compile-verified
//
#include <hip/hip_runtime.h>
#include <hip/hip_bf16.h>

typedef __attribute__((ext_vector_type(2))) float v2f;
typedef __attribute__((ext_vector_type(8))) float v8f;

namespace {
constexpr int kT      = 8192;           // time steps per batch
constexpr int kB      = 64;             // batches
constexpr int kL      = 64;             // DEC_LEN (template length)
constexpr int kNWin   = kT - kL;        // 8128 window starts
constexpr int kSplit  = 4;              // blocks per batch in kernel 1
constexpr int kWinPB  = kNWin / kSplit; // 2032 starts per block (16B-aligned offset)
constexpr int kTilesPB = kWinPB / 16;   // 127 tiles of 16 starts
constexpr int kXSeg   = kWinPB + kL;    // 2096 floats of x staged in LDS
constexpr int kXChunks = kXSeg / 4;     // 524 16-byte chunks
constexpr float kInf  = 3.402823466e38f;
}

// Low 32 bits of a generic pointer to __shared__ = LDS byte address (gfx1250
// LDS aperture maps addr[31:0] -> LDS address).
__device__ __forceinline__ unsigned lds_addr(const void* p) {
  return (unsigned)(uintptr_t)p;
}

// Async global -> LDS copy, 16 bytes per lane, tracked by ASYNCcnt.
__device__ __forceinline__ void async_copy_b128(unsigned lds_byte_addr,
                                                const void* gptr) {
  asm volatile("global_load_async_to_lds_b128 %0, %1, off"
               :: "v"(lds_byte_addr), "v"(gptr)
               : "memory");
}

__device__ __forceinline__ void wait_async_zero() {
  asm volatile("s_wait_asynccnt 0" ::: "memory");
}

// Pick acc[i] for runtime i in [0,8) via cndmask chain (diagonal extraction).
__device__ __forceinline__ float diag_pick(v8f a, int i) {
  float d = a[0];
  d = (i == 1) ? a[1] : d;
  d = (i == 2) ? a[2] : d;
  d = (i == 3) ? a[3] : d;
  d = (i == 4) ? a[4] : d;
  d = (i == 5) ? a[5] : d;
  d = (i == 6) ? a[6] : d;
  d = (i == 7) ? a[7] : d;
  return d;
}

// Kernel 1: per (batch, segment) block, compute best (dist, start) over its
// 2032 window starts using f32 WMMA Gram diagonals.
__global__ __launch_bounds__(256) void mse_argmin_kernel(
    const float* __restrict__ X_in,   // [B, T]
    const float* __restrict__ X_out,  // [B, L]
    float* __restrict__ bestD,        // [B*kSplit]
    int* __restrict__ bestS)          // [B*kSplit]
{
  __shared__ __align__(16) float sx[kXSeg];
  __shared__ __align__(16) float sy[kL];
  __shared__ float wD[8];
  __shared__ int   wS[8];

  const int blk = blockIdx.x;
  const int b   = blk / kSplit;
  const int seg = blk % kSplit;
  const int w0  = seg * kWinPB;       // first global window start of this block

  const float* x = X_in  + (size_t)b * kT;
  const float* y = X_out + (size_t)b * kL;

  // Stage x segment + y template into LDS via async global->LDS B128 copies
  // (CDNA5 path: bypasses VGPRs, tracked with ASYNCcnt).
  for (int c = threadIdx.x; c < kXChunks; c += 256) {
    async_copy_b128(lds_addr(&sx[c * 4]), x + w0 + c * 4);
  }
  if (threadIdx.x < kL / 4) {
    async_copy_b128(lds_addr(&sy[threadIdx.x * 4]), y + threadIdx.x * 4);
  }
  wait_async_zero();
  __syncthreads();

  const int lane = threadIdx.x & 31;
  const int wave = threadIdx.x >> 5;
  const int row  = lane & 15;          // matrix row this lane feeds (A layout)
  const int koff = (lane >> 4) << 1;   // lanes 0-15 -> K{0,1}, lanes 16-31 -> K{2,3}

  // Hoist the template values this lane needs (K = 4i+koff, 4i+koff+1).
  float y0[16], y1[16];
#pragma unroll
  for (int i = 0; i < 16; ++i) {
    y0[i] = sy[4 * i + koff];
    y1[i] = sy[4 * i + koff + 1];
  }

  float myBestD = kInf;
  int   myBestS = 0x7fffffff;

  for (int t = wave; t < kTilesPB; t += 8) {
    const int s0   = t * 16;           // local start base of this 16-start tile
    const int base = s0 + row;
    v8f acc = {};
#pragma unroll
    for (int i = 0; i < 16; ++i) {     // K = 64 in steps of 4
      const int k = 4 * i + koff;
      v2f a;
      a.x = sx[base + k]     - y0[i];
      a.y = sx[base + k + 1] - y1[i];
      // Gram: A and A^T share the same register layout for 16x16x4 f32,
      // so pass the same diffs as both A and B. Diagonal = sum of squares.
      acc = __builtin_amdgcn_wmma_f32_16x16x4_f32(
          false, a, false, a, (short)0, acc, false, false);
    }
    // Diagonal lives in lanes 0-7 (rows 0-7) and lanes 24-31 (rows 8-15).
    float d = kInf;
    int   s = 0x7fffffff;
    if (lane < 8) {
      d = diag_pick(acc, lane);
      s = w0 + s0 + lane;
    } else if (lane >= 24) {
      d = diag_pick(acc, lane - 24);
      s = w0 + s0 + (lane - 16);
    }
    if (d < myBestD || (d == myBestD && s < myBestS)) { myBestD = d; myBestS = s; }
  }

  // Wave argmin reduction (wave32), first-index tie-break.
  for (int off = 16; off >= 1; off >>= 1) {
    float od = __shfl_xor(myBestD, off, 32);
    int   os = __shfl_xor(myBestS, off, 32);
    if (od < myBestD || (od == myBestD && os < myBestS)) { myBestD = od; myBestS = os; }
  }
  if (lane == 0) { wD[wave] = myBestD; wS[wave] = myBestS; }
  __syncthreads();

  if (threadIdx.x == 0) {
    float bd = wD[0]; int bs = wS[0];
    for (int i = 1; i < 8; ++i) {
      if (wD[i] < bd || (wD[i] == bd && wS[i] < bs)) { bd = wD[i]; bs = wS[i]; }
    }
    bestD[blk] = bd;
    bestS[blk] = bs;
  }
}

// Kernel 2: reduce the kSplit partials per batch and gather x[start : start+L].
__global__ __launch_bounds__(64) void argmin_gather_kernel(
    const float* __restrict__ X_in,   // [B, T]
    const float* __restrict__ bestD,  // [B*kSplit]
    const int* __restrict__ bestS,    // [B*kSplit]
    float* __restrict__ out)          // [B, L]
{
  const int b = blockIdx.x;
  __shared__ int s_start;
  if (threadIdx.x == 0) {
    float bd = bestD[b * kSplit];
    int   bs = bestS[b * kSplit];
    for (int i = 1; i < kSplit; ++i) {
      float d = bestD[b * kSplit + i];
      int   s = bestS[b * kSplit + i];
      if (d < bd || (d == bd && s < bs)) { bd = d; bs = s; }
    }
    s_start = bs;
  }
  __syncthreads();
  out[b * kL + threadIdx.x] = X_in[(size_t)b * kT + s_start + threadIdx.x];
}

extern "C" void kernel_launch(void* const* d_in, const int* in_sizes, int n_in,
                              void* d_out, int out_size, void* d_ws, size_t ws_size,
                              hipStream_t stream) {
  // setup_inputs order: feats_in(0) [unused], X_in(1), feats_out(2) [unused], X_out(3)
  const float* X_in  = (const float*)d_in[1];
  const float* X_out = (const float*)d_in[3];
  float* out = (float*)d_out;

  float* bestD = (float*)d_ws;                                      // 256 floats
  int*   bestS = (int*)((char*)d_ws + kB * kSplit * sizeof(float)); // 256 ints

  mse_argmin_kernel<<<dim3(kB * kSplit), dim3(256), 0, stream>>>(X_in, X_out, bestD, bestS);
  argmin_gather_kernel<<<dim3(kB), dim3(64), 0, stream>>>(X_in, bestD, bestS, out);
}